// GCNModelVAE_56616258896068
// MI455X (gfx1250) — compile-verified
//
#include <hip/hip_runtime.h>

typedef __attribute__((ext_vector_type(2))) float v2f;
typedef __attribute__((ext_vector_type(8))) float v8f;

#define N_NODES 8192
#define E_EDGES 262144
#define F_IN    512
#define H1      256
#define H2      128
#define H3      64
#define NOUT6   6

#define WMMA_F32(a, b, c) \
    __builtin_amdgcn_wmma_f32_16x16x4_f32(false, (a), false, (b), (short)0, (c), false, false)

// ---------------------------------------------------------------------------
// zero fill (graph-capture safe scratch init)
// ---------------------------------------------------------------------------
__global__ __launch_bounds__(256) void zero_kernel(float4* p, int n4) {
    int i = blockIdx.x * 256 + threadIdx.x;
    if (i < n4) p[i] = make_float4(0.f, 0.f, 0.f, 0.f);
}

// ---------------------------------------------------------------------------
// fp32 WMMA GEMM: C[M,N] = op(A)[M,K] @ B[K,N] (+ bias[N])
// One wave per 16x64 output strip; 4 accumulators share one A fragment.
// K is a compile-time constant -> known trip count, immediate offsets.
// Software-pipelined with a peeled epilogue: in-loop loads are unconditional
// and overlap the WMMAs of the previous k-step.
// ISA layout (V_WMMA_F32_16X16X4_F32):
//   lane l: half = l>>4, lm = l&15
//   A frag a[r] = A[m0+lm][k + 2*half + r]      (r = 0,1)
//   B frag b[r] = B[k + 2*half + r][n0 + lm]
//   C frag c[r] = C[m0 + r + 8*half][n0 + lm]   (r = 0..7)
// Requires: M%16==0, N%64==0, K%4==0, EXEC all-ones (no divergence).
// ---------------------------------------------------------------------------
template <int K, bool RELU_A, bool ADD_BIAS>
__global__ __launch_bounds__(32)
void wmma_gemm_f32(const float* __restrict__ A, const float* __restrict__ B,
                   const float* __restrict__ bias, float* __restrict__ C,
                   int N) {
    const int ntiles = N >> 6;
    const int m0 = (blockIdx.x / ntiles) << 4;
    const int n0 = (blockIdx.x % ntiles) << 6;
    const int lane = threadIdx.x;
    const int half = lane >> 4;
    const int lm   = lane & 15;

    v8f c0 = {}, c1 = {}, c2 = {}, c3 = {};
    const float* arow = A + (size_t)(m0 + lm) * K + 2 * half;   // 8B aligned
    const float* bcol = B + (size_t)(2 * half) * N + n0 + lm;

    // prologue: fragments for k = 0
    v2f a, b0, b1, b2, b3;
    a    = *(const v2f*)arow;
    b0.x = bcol[0];  b0.y = bcol[N];
    b1.x = bcol[16]; b1.y = bcol[N + 16];
    b2.x = bcol[32]; b2.y = bcol[N + 32];
    b3.x = bcol[48]; b3.y = bcol[N + 48];

    for (int k = 0; k + 4 < K; k += 4) {
        // unconditional next-step loads (overlap with this step's WMMAs)
        arow += 4;
        bcol += (size_t)4 * N;
        v2f na, nb0, nb1, nb2, nb3;
        na    = *(const v2f*)arow;
        nb0.x = bcol[0];  nb0.y = bcol[N];
        nb1.x = bcol[16]; nb1.y = bcol[N + 16];
        nb2.x = bcol[32]; nb2.y = bcol[N + 32];
        nb3.x = bcol[48]; nb3.y = bcol[N + 48];

        v2f ca = a;
        if (RELU_A) { ca.x = fmaxf(ca.x, 0.f); ca.y = fmaxf(ca.y, 0.f); }
        c0 = WMMA_F32(ca, b0, c0);
        c1 = WMMA_F32(ca, b1, c1);
        c2 = WMMA_F32(ca, b2, c2);
        c3 = WMMA_F32(ca, b3, c3);

        a = na; b0 = nb0; b1 = nb1; b2 = nb2; b3 = nb3;
    }
    // epilogue: last k-step
    if (RELU_A) { a.x = fmaxf(a.x, 0.f); a.y = fmaxf(a.y, 0.f); }
    c0 = WMMA_F32(a, b0, c0);
    c1 = WMMA_F32(a, b1, c1);
    c2 = WMMA_F32(a, b2, c2);
    c3 = WMMA_F32(a, b3, c3);

    float b0v = 0.f, b1v = 0.f, b2v = 0.f, b3v = 0.f;
    if (ADD_BIAS) {
        b0v = bias[n0 + lm];      b1v = bias[n0 + 16 + lm];
        b2v = bias[n0 + 32 + lm]; b3v = bias[n0 + 48 + lm];
    }
    float* crow = C + (size_t)(m0 + 8 * half) * N + n0 + lm;
#pragma unroll
    for (int r = 0; r < 8; ++r) {
        crow[0]  = c0[r] + b0v;
        crow[16] = c1[r] + b1v;
        crow[32] = c2[r] + b2v;
        crow[48] = c3[r] + b3v;
        crow += N;
    }
}

// ---------------------------------------------------------------------------
// fp32 WMMA for C = Z @ Z^T, Z:[Mn,64] row-major. B frag = Z[n][k..k+1] is a
// contiguous float2 load; K=64 compile-time -> full unroll. This stage is
// bound by the 256MB result store (~11us at 23.3 TB/s).
// ---------------------------------------------------------------------------
__global__ __launch_bounds__(32)
void wmma_zzt_f32(const float* __restrict__ Z, float* __restrict__ C, int Mn) {
    constexpr int Kd = H3;  // 64
    const int ntiles = Mn >> 6;
    const int m0 = (blockIdx.x / ntiles) << 4;
    const int n0 = (blockIdx.x % ntiles) << 6;
    const int lane = threadIdx.x;
    const int half = lane >> 4;
    const int lm   = lane & 15;

    v8f c0 = {}, c1 = {}, c2 = {}, c3 = {};
    const float* ap = Z + (size_t)(m0 + lm) * Kd + 2 * half;
    const float* bp = Z + (size_t)(n0 + lm) * Kd + 2 * half;

#pragma unroll
    for (int k = 0; k < Kd; k += 4) {
        v2f a  = *(const v2f*)(ap + k);
        v2f b0 = *(const v2f*)(bp + k);
        v2f b1 = *(const v2f*)(bp + 16 * Kd + k);
        v2f b2 = *(const v2f*)(bp + 32 * Kd + k);
        v2f b3 = *(const v2f*)(bp + 48 * Kd + k);
        c0 = WMMA_F32(a, b0, c0);
        c1 = WMMA_F32(a, b1, c1);
        c2 = WMMA_F32(a, b2, c2);
        c3 = WMMA_F32(a, b3, c3);
    }

    float* crow = C + (size_t)(m0 + 8 * half) * Mn + n0 + lm;
#pragma unroll
    for (int r = 0; r < 8; ++r) {
        crow[0]  = c0[r];
        crow[16] = c1[r];
        crow[32] = c2[r];
        crow[48] = c3[r];
        crow += Mn;
    }
}

// ---------------------------------------------------------------------------
// COO SpMM scatter: dst[row,:] += val * src[col,:]; float4 per thread,
// consecutive threads cover consecutive features (coalesced). Accumulators
// stay resident in the 192MB L2.
// ---------------------------------------------------------------------------
template <int D>
__global__ __launch_bounds__(256)
void spmm_scatter(const int* __restrict__ rows, const int* __restrict__ cols,
                  const float* __restrict__ vals, const float* __restrict__ src,
                  float* __restrict__ dst, int E) {
    const int chunks = D >> 2;
    int idx = blockIdx.x * 256 + threadIdx.x;
    int e = idx / chunks;
    if (e >= E) return;
    int c = (idx - e * chunks) << 2;
    int r = rows[e], cl = cols[e];
    float v = vals[e];
    float4 sv = *(const float4*)(src + (size_t)cl * D + c);
    float* d = dst + (size_t)r * D + c;
    atomicAdd(d + 0, v * sv.x);
    atomicAdd(d + 1, v * sv.y);
    atomicAdd(d + 2, v * sv.z);
    atomicAdd(d + 3, v * sv.w);
}

// dual scatter: shares index/value loads for mu0 and logvar0 paths
template <int D>
__global__ __launch_bounds__(256)
void spmm_scatter2(const int* __restrict__ rows, const int* __restrict__ cols,
                   const float* __restrict__ vals,
                   const float* __restrict__ src1, float* __restrict__ dst1,
                   const float* __restrict__ src2, float* __restrict__ dst2,
                   int E) {
    const int chunks = D >> 2;
    int idx = blockIdx.x * 256 + threadIdx.x;
    int e = idx / chunks;
    if (e >= E) return;
    int c = (idx - e * chunks) << 2;
    int r = rows[e], cl = cols[e];
    float v = vals[e];
    float4 s1 = *(const float4*)(src1 + (size_t)cl * D + c);
    float4 s2 = *(const float4*)(src2 + (size_t)cl * D + c);
    float* d1 = dst1 + (size_t)r * D + c;
    float* d2 = dst2 + (size_t)r * D + c;
    atomicAdd(d1 + 0, v * s1.x); atomicAdd(d1 + 1, v * s1.y);
    atomicAdd(d1 + 2, v * s1.z); atomicAdd(d1 + 3, v * s1.w);
    atomicAdd(d2 + 0, v * s2.x); atomicAdd(d2 + 1, v * s2.y);
    atomicAdd(d2 + 2, v * s2.z); atomicAdd(d2 + 3, v * s2.w);
}

// ---------------------------------------------------------------------------
// out6 = z @ dense1_W + dense1_b  (N=6 tail, trivial VALU work)
// ---------------------------------------------------------------------------
__global__ __launch_bounds__(256)
void out6_kernel(const float* __restrict__ Z, const float* __restrict__ W,
                 const float* __restrict__ bias, float* __restrict__ out) {
    int row = blockIdx.x * 256 + threadIdx.x;
    float acc[NOUT6];
#pragma unroll
    for (int j = 0; j < NOUT6; ++j) acc[j] = bias[j];
    const float* zr = Z + (size_t)row * H3;
    for (int k = 0; k < H3; ++k) {
        float zv = zr[k];
#pragma unroll
        for (int j = 0; j < NOUT6; ++j) acc[j] = fmaf(zv, W[k * NOUT6 + j], acc[j]);
    }
    float* o = out + (size_t)row * NOUT6;
#pragma unroll
    for (int j = 0; j < NOUT6; ++j) o[j] = acc[j];
}

// ---------------------------------------------------------------------------
extern "C" void kernel_launch(void* const* d_in, const int* in_sizes, int n_in,
                              void* d_out, int out_size, void* d_ws, size_t ws_size,
                              hipStream_t stream) {
    (void)in_sizes; (void)n_in; (void)out_size; (void)ws_size;

    const float* x    = (const float*)d_in[0];
    const int*   rows = (const int*)d_in[1];
    const int*   cols = (const int*)d_in[2];
    const float* vals = (const float*)d_in[3];
    const float* W1   = (const float*)d_in[4];
    const float* W2   = (const float*)d_in[5];
    const float* W3   = (const float*)d_in[6];
    const float* dW   = (const float*)d_in[7];
    const float* db   = (const float*)d_in[8];
    const float* d1W  = (const float*)d_in[9];
    const float* d1b  = (const float*)d_in[10];

    // workspace layout (floats): 32 MB total
    float* ws    = (float*)d_ws;
    float* xW1   = ws;                                // [8192,256]
    float* h1acc = xW1   + (size_t)N_NODES * H1;      // [8192,256] spmm accum (pre-relu)
    float* t2    = h1acc + (size_t)N_NODES * H1;      // [8192,128]
    float* t3    = t2    + (size_t)N_NODES * H2;      // [8192,128]
    float* mu0   = t3    + (size_t)N_NODES * H2;      // [8192,128]
    float* lv0   = mu0   + (size_t)N_NODES * H2;      // [8192,128]

    // output layout: adj_rec | out6 | mu | logvar
    float* outf    = (float*)d_out;
    float* adj_rec = outf;
    float* out6    = adj_rec + (size_t)N_NODES * N_NODES;
    float* mu      = out6    + (size_t)N_NODES * NOUT6;
    float* logvar  = mu      + (size_t)N_NODES * H3;

    // 1) zero scatter accumulators (h1acc; mu0+lv0 contiguous)
    {
        int n4 = (N_NODES * H1) / 4;
        zero_kernel<<<n4 / 256, 256, 0, stream>>>((float4*)h1acc, n4);
        int n4b = (2 * N_NODES * H2) / 4;
        zero_kernel<<<n4b / 256, 256, 0, stream>>>((float4*)mu0, n4b);
    }

    // 2) xW1 = x @ W1
    wmma_gemm_f32<F_IN, false, false><<<(N_NODES / 16) * (H1 / 64), 32, 0, stream>>>(
        x, W1, nullptr, xW1, H1);

    // 3) h1acc = spmm(xW1)
    {
        long total = (long)E_EDGES * (H1 / 4);
        spmm_scatter<H1><<<(int)(total / 256), 256, 0, stream>>>(
            rows, cols, vals, xW1, h1acc, E_EDGES);
    }

    // 4) t2 = relu(h1acc) @ W2 ; t3 = relu(h1acc) @ W3
    wmma_gemm_f32<H1, true, false><<<(N_NODES / 16) * (H2 / 64), 32, 0, stream>>>(
        h1acc, W2, nullptr, t2, H2);
    wmma_gemm_f32<H1, true, false><<<(N_NODES / 16) * (H2 / 64), 32, 0, stream>>>(
        h1acc, W3, nullptr, t3, H2);

    // 5) mu0 = spmm(t2) ; lv0 = spmm(t3)  (fused dual scatter)
    {
        long total = (long)E_EDGES * (H2 / 4);
        spmm_scatter2<H2><<<(int)(total / 256), 256, 0, stream>>>(
            rows, cols, vals, t2, mu0, t3, lv0, E_EDGES);
    }

    // 6) mu = mu0 @ dense_W + b ; logvar = lv0 @ dense_W + b  (into d_out)
    wmma_gemm_f32<H2, false, true><<<(N_NODES / 16) * (H3 / 64), 32, 0, stream>>>(
        mu0, dW, db, mu, H3);
    wmma_gemm_f32<H2, false, true><<<(N_NODES / 16) * (H3 / 64), 32, 0, stream>>>(
        lv0, dW, db, logvar, H3);

    // 7) adj_rec = mu @ mu^T (dominant: 256MB store, HBM-bound)
    wmma_zzt_f32<<<(N_NODES / 16) * (N_NODES / 64), 32, 0, stream>>>(
        mu, adj_rec, N_NODES);

    // 8) out6 = mu @ dense1_W + dense1_b
    out6_kernel<<<N_NODES / 256, 256, 0, stream>>>(mu, d1W, d1b, out6);
}